// DeformableConv2d_19816979103771
// MI455X (gfx1250) — compile-verified
//
#include <hip/hip_runtime.h>
#include <hip/hip_bf16.h>

#define BB 4
#define CC 32
#define HH 128
#define WWID 256
#define OO 64
#define HWSZ (HH * WWID)      // 32768
#define K2 9
#define KCONV (CC * K2)       // 288
#define MPAD 32               // 18 offset + 9 modulator channels, padded to 32

// conv x-patch LDS: channel-interleaved-by-4 layout:
//   addr(c,row,col) = (c>>2)*XGRP + row*XROW + (col<<2) + (c&3)
#define XPITCH 132            // padded patch row (130 cols used)
#define XCSTR  (3 * XPITCH)   // 396 elements per channel (staging index space)
#define XROW   (XPITCH * 4)   // 528
#define XGRP   (3 * XROW)     // 1584
#define APITCH 292            // conv weight row pitch (288 used; 292%64=36, gcd(36,64)=4 -> conflict-free b64)
#define PBP    520            // pointwise B group pitch (520%64=8 -> conflict-free)
#define WPITCH 36             // pointwise W row pitch
#define TPITCH 133            // transpose tile pitch (odd -> conflict-free both phases)

typedef __attribute__((ext_vector_type(2))) float v2f;
typedef __attribute__((ext_vector_type(4))) float v4f;
typedef __attribute__((ext_vector_type(8))) float v8f;

static __device__ __forceinline__ v4f v4splat(float f) {
    v4f r; r.x = f; r.y = f; r.z = f; r.w = f; return r;
}

// ---------------------------------------------------------------------------
// Kernel 0: pack w_off (18,32,3,3) + w_mod (9,32,3,3) into padded [32][288]
// with K reordered as kk = (ky*3+kx)*32 + c.
// ---------------------------------------------------------------------------
__global__ void pack_w(const float* __restrict__ w_off,
                       const float* __restrict__ w_mod,
                       float* __restrict__ wpad) {
    int i = blockIdx.x * 256 + threadIdx.x;
    if (i >= MPAD * KCONV) return;
    int m = i / KCONV, t = i - m * KCONV;
    int r = t >> 5, c = t & 31;            // kk = r*32 + c, r = ky*3+kx
    float v = 0.0f;
    if (m < 18)      v = w_off[m * KCONV + c * K2 + r];
    else if (m < 27) v = w_mod[(m - 18) * KCONV + c * K2 + r];
    wpad[i] = v;
}

// ---------------------------------------------------------------------------
// Kernel 1: NCHW -> NHWC transpose of x through LDS (both phases coalesced /
// bank-conflict-free). Makes the deformable gather 16B-per-lane instead of
// 4B-per-lane and cuts its VMEM issue count ~4x.
// ---------------------------------------------------------------------------
__global__ void __launch_bounds__(256)
transpose_x(const float* __restrict__ x, float* __restrict__ xT) {
    __shared__ float sT[CC * TPITCH];
    const int tid = threadIdx.x;
    int blk = blockIdx.x;                 // B*HW/128 = 1024 blocks
    int b   = blk >> 8;
    int n0  = (blk & 255) * 128;
    const float* xb = x + (size_t)b * CC * HWSZ + n0;
    for (int i = tid; i < CC * 128; i += 256) {
        int c = i >> 7, col = i & 127;
        sT[c * TPITCH + col] = xb[(size_t)c * HWSZ + col];
    }
    __syncthreads();
    float* ob = xT + ((size_t)b * HWSZ + n0) * CC;
    for (int i = tid; i < CC * 128; i += 256) {
        int col = i >> 5, c = i & 31;     // i == col*32 + c -> ob[i] coalesced
        ob[i] = sT[c * TPITCH + col];
    }
}

// ---------------------------------------------------------------------------
// Kernel 2: offset+modulator conv as LDS-staged implicit GEMM.
// Block = 256 thr (8 waves) = 128 consecutive pixels of one image row.
// Channel-interleaved patch => each B-frag is ONE aligned ds_load_b64,
// conflict-free. Each wave: 32(M) x 16(N), K=288 unrolled -> 144 WMMAs.
// ---------------------------------------------------------------------------
__global__ void __launch_bounds__(256)
conv_offmod_wmma(const float* __restrict__ x,
                 const float* __restrict__ wpad,
                 const float* __restrict__ b_off,
                 const float* __restrict__ b_mod,
                 float* __restrict__ offs,
                 float* __restrict__ mod) {
    __shared__ alignas(16) float sX[8 * XGRP];        // 12672 floats (~50 KB)
    __shared__ alignas(16) float sA[MPAD * APITCH];   //  9344 floats (~37 KB)

    const int tid  = threadIdx.x;
    const int lane = tid & 31;
    const int wv   = tid >> 5;
    const int l16  = lane & 15;
    const int koff = (lane < 16) ? 0 : 2;

    int blk = blockIdx.x;                 // 1024 blocks
    int b   = blk >> 8;
    int n0  = (blk & 255) * 128;
    int h   = n0 >> 8;
    int w0  = n0 & 255;

    const float* xb = x + (size_t)b * CC * HWSZ;

    // ---- stage packed weights ----
    for (int i = tid; i < MPAD * KCONV; i += 256) {
        int m = i / KCONV, kk = i - m * KCONV;
        sA[m * APITCH + kk] = wpad[i];
    }
    // ---- stage x patch (zero-filled halo), channel-interleaved-by-4 ----
    for (int i = tid; i < CC * XCSTR; i += 256) {
        int c    = i / XCSTR;
        int remm = i - c * XCSTR;
        int row  = remm / XPITCH;
        int col  = remm - row * XPITCH;
        int gy   = h - 1 + row;
        int gx   = w0 - 1 + col;
        float v  = 0.0f;
        if (col < 130 && (unsigned)gy < (unsigned)HH && (unsigned)gx < (unsigned)WWID)
            v = xb[c * HWSZ + gy * WWID + gx];
        sX[(c >> 2) * XGRP + row * XROW + (col << 2) + (c & 3)] = v;
    }
    __syncthreads();

    const int p = wv * 16 + l16;          // this lane's local pixel (N index)
    v8f acc0 = {}, acc1 = {};
#pragma unroll
    for (int r = 0; r < K2; ++r) {
        const int ky = r / 3, kx = r - (r / 3) * 3;   // compile-time consts
        const int xoff = ky * XROW + ((p + kx) << 2) + koff;
        const int kbase = r * 32 + koff;
#pragma unroll
        for (int cs = 0; cs < 8; ++cs) {
            v2f a0 = *(const v2f*)&sA[l16 * APITCH + kbase + cs * 4];
            v2f a1 = *(const v2f*)&sA[(16 + l16) * APITCH + kbase + cs * 4];
            v2f bf = *(const v2f*)&sX[cs * XGRP + xoff];
            acc0 = __builtin_amdgcn_wmma_f32_16x16x4_f32(
                       false, a0, false, bf, (short)0, acc0, false, false);
            acc1 = __builtin_amdgcn_wmma_f32_16x16x4_f32(
                       false, a1, false, bf, (short)0, acc1, false, false);
        }
    }

    // ---- epilogue: bias; ch 0..17 -> offsets, 18..26 -> 2*sigmoid ----
    int nOut = n0 + p;
    int mb   = (lane < 16) ? 0 : 8;
#pragma unroll
    for (int j = 0; j < 8; ++j) {
        int ch0 = mb + j;                              // 0..15: always offset
        offs[((size_t)b * 18 + ch0) * HWSZ + nOut] = acc0[j] + b_off[ch0];
        int ch1 = 16 + mb + j;                         // 16..31
        if (ch1 < 18) {
            offs[((size_t)b * 18 + ch1) * HWSZ + nOut] = acc1[j] + b_off[ch1];
        } else if (ch1 < 27) {
            float s = acc1[j] + b_mod[ch1 - 18];
            mod[((size_t)b * 9 + (ch1 - 18)) * HWSZ + nOut] =
                2.0f / (1.0f + __expf(-s));
        }
    }
}

// ---------------------------------------------------------------------------
// Kernel 3: deformable bilinear sampling + modulated depthwise reduction.
// Thread per (b,h,w). NHWC x => each corner = 8 contiguous b128 loads.
// Output dwT is NHWC (contiguous b128 stores).
// ---------------------------------------------------------------------------
__global__ void __launch_bounds__(256)
deform_dw(const float* __restrict__ xT,
          const float* __restrict__ offs,
          const float* __restrict__ mod,
          const float* __restrict__ w_dw,
          float* __restrict__ dwT) {
    __shared__ alignas(16) float sdwT[K2 * CC];       // transposed [k][c]
    for (int i = threadIdx.x; i < K2 * CC; i += 256) {
        int k = i >> 5, c = i & 31;
        sdwT[i] = w_dw[c * K2 + k];
    }
    __syncthreads();

    long idx = (long)blockIdx.x * blockDim.x + threadIdx.x;   // B*HW exact
    int b = (int)(idx / HWSZ);
    int n = (int)(idx % HWSZ);
    int h = n >> 8, w = n & 255;
    const float* xbT = xT + (size_t)b * HWSZ * CC;
    const float* ofb = offs + (size_t)b * 18 * HWSZ + n;
    const float* mdb = mod + (size_t)b * 9 * HWSZ + n;

    v4f acc4[8];
#pragma unroll
    for (int cc = 0; cc < 8; ++cc) acc4[cc] = v4splat(0.0f);

#pragma unroll
    for (int k = 0; k < K2; ++k) {
        float dy = ofb[(size_t)(2 * k) * HWSZ];
        float dx = ofb[(size_t)(2 * k + 1) * HWSZ];
        float mk = mdb[(size_t)k * HWSZ];

        float py = (float)(h - 1 + k / 3) + dy;
        float px = (float)(w - 1 + k % 3) + dx;
        float y0f = floorf(py), x0f = floorf(px);
        int y0 = (int)y0f, x0 = (int)x0f;
        float wy1 = py - y0f, wx1 = px - x0f;
        float wy0 = 1.0f - wy1, wx0 = 1.0f - wx1;
        int y1 = y0 + 1, x1 = x0 + 1;

        bool vy0 = (unsigned)y0 < (unsigned)HH, vy1 = (unsigned)y1 < (unsigned)HH;
        bool vx0 = (unsigned)x0 < (unsigned)WWID, vx1 = (unsigned)x1 < (unsigned)WWID;
        v4f w00 = v4splat(wy0 * wx0 * (float)(vy0 && vx0));
        v4f w01 = v4splat(wy0 * wx1 * (float)(vy0 && vx1));
        v4f w10 = v4splat(wy1 * wx0 * (float)(vy1 && vx0));
        v4f w11 = v4splat(wy1 * wx1 * (float)(vy1 && vx1));

        int yc0 = min(max(y0, 0), HH - 1),   yc1 = min(max(y1, 0), HH - 1);
        int xc0 = min(max(x0, 0), WWID - 1), xc1 = min(max(x1, 0), WWID - 1);
        const v4f* q00 = (const v4f*)(xbT + (size_t)(yc0 * WWID + xc0) * CC);
        const v4f* q01 = (const v4f*)(xbT + (size_t)(yc0 * WWID + xc1) * CC);
        const v4f* q10 = (const v4f*)(xbT + (size_t)(yc1 * WWID + xc0) * CC);
        const v4f* q11 = (const v4f*)(xbT + (size_t)(yc1 * WWID + xc1) * CC);
        const v4f* wd  = (const v4f*)(sdwT + k * CC);
        v4f mkv = v4splat(mk);
#pragma unroll
        for (int cc = 0; cc < 8; ++cc) {
            v4f s = q00[cc] * w00 + q01[cc] * w01 + q10[cc] * w10 + q11[cc] * w11;
            acc4[cc] = acc4[cc] + (mkv * wd[cc]) * s;
        }
    }
    v4f* o = (v4f*)(dwT + ((size_t)b * HWSZ + n) * CC);
#pragma unroll
    for (int cc = 0; cc < 8; ++cc) o[cc] = acc4[cc];
}

// ---------------------------------------------------------------------------
// Kernel 4: pointwise conv as LDS-staged GEMM (M=64, K=32, N=128/block).
// NHWC input => staging is a straight sequential copy; channel-interleaved
// tile => one ds_load_b64 per B-frag, conflict-free. 32 WMMAs per wave.
// ---------------------------------------------------------------------------
__global__ void __launch_bounds__(256)
pointwise_wmma(const float* __restrict__ dwT,
               const float* __restrict__ w_pw,
               float* __restrict__ out) {
    __shared__ alignas(16) float sB[8 * PBP];         // 4160 floats
    __shared__ alignas(16) float sW[OO * WPITCH];     // 2304 floats

    const int tid  = threadIdx.x;
    const int lane = tid & 31;
    const int wv   = tid >> 5;
    const int l16  = lane & 15;
    const int koff = (lane < 16) ? 0 : 2;

    int blk = blockIdx.x;                 // 1024 blocks
    int b   = blk >> 8;
    int n0  = (blk & 255) * 128;
    const float* db = dwT + ((size_t)b * HWSZ + n0) * CC;

    for (int i = tid; i < OO * CC; i += 256) {
        int m = i >> 5, c = i & 31;
        sW[m * WPITCH + c] = w_pw[i];
    }
    for (int i = tid; i < CC * 128; i += 256) {       // i = col*32 + c
        int col = i >> 5, c = i & 31;
        sB[(c >> 2) * PBP + (col << 2) + (c & 3)] = db[i];
    }
    __syncthreads();

    const int p = wv * 16 + l16;
    v8f acc[4] = {{}, {}, {}, {}};
#pragma unroll
    for (int ks = 0; ks < 8; ++ks) {
        v2f bf = *(const v2f*)&sB[ks * PBP + (p << 2) + koff];
#pragma unroll
        for (int mt = 0; mt < 4; ++mt) {
            v2f a = *(const v2f*)&sW[(mt * 16 + l16) * WPITCH + ks * 4 + koff];
            acc[mt] = __builtin_amdgcn_wmma_f32_16x16x4_f32(
                          false, a, false, bf, (short)0, acc[mt], false, false);
        }
    }

    int nOut = n0 + p;
    int mb   = (lane < 16) ? 0 : 8;
#pragma unroll
    for (int mt = 0; mt < 4; ++mt)
#pragma unroll
        for (int j = 0; j < 8; ++j)
            out[((size_t)b * OO + mt * 16 + mb + j) * HWSZ + nOut] = acc[mt][j];
}

// ---------------------------------------------------------------------------
extern "C" void kernel_launch(void* const* d_in, const int* in_sizes, int n_in,
                              void* d_out, int out_size, void* d_ws, size_t ws_size,
                              hipStream_t stream) {
    const float* x     = (const float*)d_in[0];
    const float* w_off = (const float*)d_in[1];
    const float* b_off = (const float*)d_in[2];
    const float* w_mod = (const float*)d_in[3];
    const float* b_mod = (const float*)d_in[4];
    const float* w_dw  = (const float*)d_in[5];
    const float* w_pw  = (const float*)d_in[6];
    float* out = (float*)d_out;

    float* ws   = (float*)d_ws;
    float* offs = ws;                               // B*18*HW floats
    float* mod  = offs + (size_t)BB * 18 * HWSZ;    // B*9*HW
    float* dwT  = mod  + (size_t)BB * 9  * HWSZ;    // B*HW*C (NHWC)
    float* xT   = dwT  + (size_t)BB * CC * HWSZ;    // B*HW*C (NHWC)
    float* wpad = xT   + (size_t)BB * CC * HWSZ;    // 32*288

    pack_w<<<(MPAD * KCONV + 255) / 256, 256, 0, stream>>>(w_off, w_mod, wpad);

    transpose_x<<<(BB * HWSZ) / 128, 256, 0, stream>>>(x, xT);

    conv_offmod_wmma<<<(BB * HWSZ) / 128, 256, 0, stream>>>(
        x, wpad, b_off, b_mod, offs, mod);

    deform_dw<<<(BB * HWSZ) / 256, 256, 0, stream>>>(xT, offs, mod, w_dw, dwT);

    pointwise_wmma<<<(BB * HWSZ) / 128, 256, 0, stream>>>(dwT, w_pw, out);
}